// S4DLayer_1864015806523
// MI455X (gfx1250) — compile-verified
//
#include <hip/hip_runtime.h>
#include <hip/hip_bf16.h>

// S4D layer, chunked-scan (SSD) formulation for MI455X / gfx1250.
//   H=512 channels, N/2=32 complex states, B=8, L=4096, chunk Q=64, NC=64.
// Per (b,h) workgroup:  W = U@P  (WMMA f32) -> 64-step state scan ->
//                       Y = U@T + S@M (WMMA f32, fused accumulate) + D*u.

#define H_DIM  512
#define N2     32
#define BATCHN 8
#define LSEQ   4096
#define Q      64
#define NC     64
#define LDP    65            // padded LDS leading dim (bank-conflict free)

// per-h matrix block in workspace (floats)
#define OFF_P   0            // P[t][j]   : input -> state      (64x64)
#define OFF_T   4096         // T[m][t]   : Toeplitz K[t-m]     (64x64)
#define OFF_M   8192         // M[j][t]   : state -> output     (64x64)
#define OFF_AQ  12288        // dA^Q re/im interleaved          (64)
#define MAT_STRIDE 12352     // floats per h  (512*12352*4B ~= 25.3 MB ws)

typedef float v2f __attribute__((ext_vector_type(2)));
typedef float v8f __attribute__((ext_vector_type(8)));

__device__ __forceinline__ v8f wmma4(v2f a, v2f b, v8f c) {
  // D(16x16,f32) = A(16x4,f32) * B(4x16,f32) + C
  return __builtin_amdgcn_wmma_f32_16x16x4_f32(
      /*neg_a=*/false, a, /*neg_b=*/false, b,
      /*c_mod=*/(short)0, c, /*reuse_a=*/false, /*reuse_b=*/false);
}

// ---------------------------------------------------------------------------
// Setup: per h, bilinear discretization + Vandermonde-derived matrices.
// grid = 512 blocks, 32 lanes (one lane per complex state n).
// ---------------------------------------------------------------------------
__global__ void __launch_bounds__(32)
s4d_setup(const float* __restrict__ log_delta,
          const float* __restrict__ log_A_real,
          const float* __restrict__ A_imag,
          const float* __restrict__ Bmat,
          const float* __restrict__ Cmat,
          float* __restrict__ ws)
{
  __shared__ float kpart[Q][N2];
  __shared__ float kv[Q];
  const int h = blockIdx.x;
  const int n = threadIdx.x;                 // 0..31
  float* __restrict__ mh = ws + (size_t)h * MAT_STRIDE;

  const float delta = expf(log_delta[h]);    // log_delta is (H,1)
  const int hn = h * N2 + n;
  const float Ar = -expf(log_A_real[hn]);
  const float Ai = A_imag[hn];
  // dA = (1 + delta*A/2) / (1 - delta*A/2)
  const float zr = 0.5f * delta * Ar;
  const float zi = 0.5f * delta * Ai;
  const float dr = 1.0f - zr, di = -zi;
  const float inv = 1.0f / (dr * dr + di * di);
  const float nr = 1.0f + zr, nim = zi;
  const float dAr = (nr * dr + nim * di) * inv;
  const float dAi = (nim * dr - nr * di) * inv;
  const float Bv  = Bmat[hn];
  const float BCr = Bv * Cmat[2 * hn + 0];
  const float BCi = Bv * Cmat[2 * hn + 1];

  float pr = 1.0f, pi = 0.0f;                // p = dA^t
  for (int t = 0; t < Q; ++t) {
    kpart[t][n] = BCr * pr - BCi * pi;       // Re(BC * dA^t), per lane
    // P[t'][2n],P[t'][2n+1] = re/im dA^(Q-1-t')  ->  row (Q-1-t) at exponent t
    mh[OFF_P + (Q - 1 - t) * Q + 2 * n]     = pr;
    mh[OFF_P + (Q - 1 - t) * Q + 2 * n + 1] = pi;
    const float qr = pr * dAr - pi * dAi;    // q = dA^(t+1)
    const float qi = pr * dAi + pi * dAr;
    // cross-chunk read-out:  y_cross[t] = sum_j S[j]*M[j][t]
    mh[OFF_M + (2 * n)     * Q + t] =  2.0f * (BCr * qr - BCi * qi);
    mh[OFF_M + (2 * n + 1) * Q + t] = -2.0f * (BCr * qi + BCi * qr);
    pr = qr; pi = qi;
  }
  mh[OFF_AQ + 2 * n]     = pr;               // dA^Q
  mh[OFF_AQ + 2 * n + 1] = pi;
  __syncthreads();

  // K[t] = 2 * sum_n Re(BC * dA^t)
  for (int t = n; t < Q; t += N2) {
    float s = 0.0f;
#pragma unroll
    for (int j = 0; j < N2; ++j) s += kpart[t][j];
    kv[t] = 2.0f * s;
  }
  __syncthreads();

  // T[m][t] = K[t-m] for t>=m else 0  (so intra = U_chunk @ T)
  for (int i = n; i < Q * Q; i += N2) {
    const int m = i >> 6, t = i & 63;
    mh[OFF_T + m * Q + t] = (t >= m) ? kv[t - m] : 0.0f;
  }
}

// ---------------------------------------------------------------------------
// Main: one workgroup (4 wave32s) per (b,h) sequence.
// ---------------------------------------------------------------------------
__global__ void __launch_bounds__(128)
s4d_main(const float* __restrict__ u,
         const float* __restrict__ Dvec,
         const float* __restrict__ ws,
         float* __restrict__ y)
{
  __shared__ float Ul[NC * LDP];   // u chunks      [c][t]
  __shared__ float Wl[NC * LDP];   // chunk inputs  [c][j]
  __shared__ float Sl[NC * LDP];   // chunk states  [c][j]
  // total static LDS: 3 * 64*65 * 4B = 49,920 B

  const int blk  = blockIdx.x;
  const int b    = blk >> 9;              // / H_DIM
  const int h    = blk & (H_DIM - 1);
  const int tid  = threadIdx.x;
  const int wv   = tid >> 5;
  const int lane = tid & 31;
  const int l15  = lane & 15;
  const int hi   = lane >> 4;
  const int m0   = wv * 16;               // this wave's 16-row band
  const int koff = hi ? 2 : 0;            // A/B frag K offset for upper half-wave

  const float* __restrict__ mh   = ws + (size_t)h * MAT_STRIDE;
  const float* __restrict__ ucol = u + (size_t)b * LSEQ * H_DIM + h;

  // stage this sequence's u column into LDS as (chunk, t)
  for (int i = tid; i < LSEQ; i += 128)
    Ul[(i >> 6) * LDP + (i & 63)] = ucol[(size_t)i * H_DIM];
  __syncthreads();

  // ---- GEMM 1:  W = U @ P   (64x64x64, fp32 WMMA) ----
  for (int ni = 0; ni < 4; ++ni) {
    v8f acc = {};
    const int col = ni * 16 + l15;
#pragma unroll
    for (int kk = 0; kk < 16; ++kk) {
      const int kb = kk * 4 + koff;
      v2f a, bf;
      a.x  = Ul[(m0 + l15) * LDP + kb];
      a.y  = Ul[(m0 + l15) * LDP + kb + 1];
      bf.x = mh[OFF_P + kb * Q + col];        // B frags from L2-hot workspace
      bf.y = mh[OFF_P + (kb + 1) * Q + col];
      acc  = wmma4(a, bf, acc);
    }
#pragma unroll
    for (int r = 0; r < 8; ++r)
      Wl[(m0 + r + hi * 8) * LDP + col] = acc[r];
  }
  __syncthreads();

  // ---- chunk-state scan (wave 0): s_{c+1} = dA^Q * s_c + w_c ----
  if (tid < N2) {
    const float aqr = mh[OFF_AQ + 2 * tid];
    const float aqi = mh[OFF_AQ + 2 * tid + 1];
    float sr = 0.0f, si = 0.0f;
    for (int c = 0; c < NC; ++c) {
      Sl[c * LDP + 2 * tid]     = sr;         // state *entering* chunk c
      Sl[c * LDP + 2 * tid + 1] = si;
      const float wr = Wl[c * LDP + 2 * tid];
      const float wi = Wl[c * LDP + 2 * tid + 1];
      const float t0 = aqr * sr - aqi * si + wr;
      si = aqr * si + aqi * sr + wi;
      sr = t0;
    }
  }
  __syncthreads();

  // ---- GEMM 2+3 fused:  Y = U @ T + S @ M, then + D*u, store ----
  const float Dv = Dvec[h];
  float* __restrict__ ycol = y + (size_t)b * LSEQ * H_DIM + h;
  for (int ni = 0; ni < 4; ++ni) {
    v8f acc = {};
    const int col = ni * 16 + l15;
#pragma unroll
    for (int kk = 0; kk < 16; ++kk) {         // intra-chunk Toeplitz conv
      const int kb = kk * 4 + koff;
      v2f a, bf;
      a.x  = Ul[(m0 + l15) * LDP + kb];
      a.y  = Ul[(m0 + l15) * LDP + kb + 1];
      bf.x = mh[OFF_T + kb * Q + col];
      bf.y = mh[OFF_T + (kb + 1) * Q + col];
      acc  = wmma4(a, bf, acc);
    }
#pragma unroll
    for (int kk = 0; kk < 16; ++kk) {         // cross-chunk state read-out
      const int kb = kk * 4 + koff;
      v2f a, bf;
      a.x  = Sl[(m0 + l15) * LDP + kb];
      a.y  = Sl[(m0 + l15) * LDP + kb + 1];
      bf.x = mh[OFF_M + kb * Q + col];
      bf.y = mh[OFF_M + (kb + 1) * Q + col];
      acc  = wmma4(a, bf, acc);
    }
#pragma unroll
    for (int r = 0; r < 8; ++r) {
      const int c = m0 + r + hi * 8;
      const int l = c * Q + col;
      ycol[(size_t)l * H_DIM] = acc[r] + Dv * Ul[c * LDP + col];
    }
  }
}

// ---------------------------------------------------------------------------
extern "C" void kernel_launch(void* const* d_in, const int* in_sizes, int n_in,
                              void* d_out, int out_size, void* d_ws, size_t ws_size,
                              hipStream_t stream) {
  const float* u   = (const float*)d_in[0];  // (8,4096,512)
  const float* ld  = (const float*)d_in[1];  // (512,1)
  const float* lar = (const float*)d_in[2];  // (512,32)
  const float* ai  = (const float*)d_in[3];  // (512,32)
  const float* Bm  = (const float*)d_in[4];  // (512,32)
  const float* Cm  = (const float*)d_in[5];  // (512,32,2)
  const float* Dm  = (const float*)d_in[6];  // (512,)
  float* yout = (float*)d_out;               // (8,4096,512)
  float* ws   = (float*)d_ws;                // needs ~25.3 MB

  s4d_setup<<<H_DIM, 32, 0, stream>>>(ld, lar, ai, Bm, Cm, ws);
  s4d_main<<<BATCHN * H_DIM, 128, 0, stream>>>(u, Dm, ws, yout);
}